// TemporalContrastiveLoss_49117245997276
// MI455X (gfx1250) — compile-verified
//
#include <hip/hip_runtime.h>

typedef __attribute__((ext_vector_type(2))) float v2f;
typedef __attribute__((ext_vector_type(8))) float v8f;

#define BDIM 16
#define SDIM 4096
#define DDIM 512
#define TGAP 10
#define MARGIN 0.5f
#define EPS 1e-8f
// count = sum_{k=10}^{19} (S-k) = 10*4096 - 145
#define DIST_COUNT 40815.0f

__device__ __forceinline__ float waveReduceSum(float v) {
    #pragma unroll
    for (int m = 16; m > 0; m >>= 1)
        v += __shfl_xor(v, m, 32);
    return v;
}

// Kernel 1: per-row L2 norms into d_ws; block 0 thread 0 zeroes the output scalar.
__global__ __launch_bounds__(256) void tcl_norms_kernel(const float* __restrict__ h,
                                                        float* __restrict__ norms,
                                                        float* __restrict__ out) {
    if (blockIdx.x == 0 && threadIdx.x == 0) out[0] = 0.0f;
    const int lane = threadIdx.x & 31;
    const int wave = threadIdx.x >> 5;
    const int row  = blockIdx.x * (blockDim.x >> 5) + wave;   // 0 .. B*S-1
    const float4* r = (const float4*)(h + (size_t)row * DDIM);
    float s = 0.0f;
    #pragma unroll
    for (int i = lane; i < DDIM / 4; i += 32) {
        float4 v = r[i];
        s += v.x * v.x + v.y * v.y + v.z * v.z + v.w * v.w;
    }
    s = waveReduceSum(s);
    if (lane == 0) norms[row] = sqrtf(s);
}

__device__ __forceinline__ void tcl_extract(const v8f& acc, int c, int t0, int half,
                                            const float* __restrict__ nb,
                                            float& adj, float& dist) {
    if (c >= SDIM) return;                 // after all WMMAs; divergence OK here
    const float nc = nb[c];
    #pragma unroll
    for (int v = 0; v < 8; ++v) {
        const int t = t0 + v + 8 * half;   // C/D layout: M = v + 8*(lane>>4)
        const int k = c - t;
        const bool isAdj  = (k == 1);
        const bool isDist = (k >= TGAP) && (k < 2 * TGAP);
        if (!isAdj && !isDist) continue;
        const float denom = fmaxf(nb[t] * nc, EPS);
        const float sim   = acc[v] / denom;
        if (isAdj) adj  += 1.0f - sim;
        else       dist += fmaxf(MARGIN - sim, 0.0f);
    }
}

// Kernel 2: one wave per 16-row tile. Three Gram blocks A(16xD) x B_j(Dx16) via
// v_wmma_f32_16x16x4_f32 chains, extract needed diagonals, atomicAdd scalar.
__global__ __launch_bounds__(32) void tcl_gram_kernel(const float* __restrict__ h,
                                                      const float* __restrict__ norms,
                                                      float* __restrict__ out) {
    const int tilesPerBatch = SDIM / 16;                // 256
    const int b  = blockIdx.x / tilesPerBatch;
    const int t0 = (blockIdx.x % tilesPerBatch) * 16;

    const int lane = threadIdx.x;        // 0..31, EXEC all ones for WMMA
    const int mn   = lane & 15;          // M for A-frag, N for B-frag
    const int half = lane >> 4;
    const int koff = 2 * half;           // A/B 16x4 layout: VGPR0=K0/K2, VGPR1=K1/K3

    const float* base = h + (size_t)b * SDIM * DDIM;
    const float* rowA = base + (size_t)(t0 + mn) * DDIM;

    const int c0 = t0 + mn;
    const int c1 = t0 + 16 + mn;
    const int c2 = t0 + 32 + mn;
    const bool ok1 = (c1 < SDIM);
    const bool ok2 = (c2 < SDIM);
    const float* rowB0 = base + (size_t)c0 * DDIM;                      // always in range
    const float* rowB1 = base + (size_t)(ok1 ? c1 : SDIM - 1) * DDIM;   // clamped
    const float* rowB2 = base + (size_t)(ok2 ? c2 : SDIM - 1) * DDIM;   // clamped

    const v2f vzero = {0.0f, 0.0f};
    v8f acc0 = {}, acc1 = {}, acc2 = {};

    #pragma unroll 4
    for (int kk = 0; kk < DDIM; kk += 4) {
        const v2f a   = *(const v2f*)(rowA  + kk + koff);
        const v2f b0  = *(const v2f*)(rowB0 + kk + koff);
        const v2f b1r = *(const v2f*)(rowB1 + kk + koff);
        const v2f b2r = *(const v2f*)(rowB2 + kk + koff);
        const v2f b1  = ok1 ? b1r : vzero;   // branchless select, EXEC stays full
        const v2f b2  = ok2 ? b2r : vzero;

        acc0 = __builtin_amdgcn_wmma_f32_16x16x4_f32(false, a, false, b0,
                                                     (short)0, acc0, false, false);
        acc1 = __builtin_amdgcn_wmma_f32_16x16x4_f32(false, a, false, b1,
                                                     (short)0, acc1, false, false);
        acc2 = __builtin_amdgcn_wmma_f32_16x16x4_f32(false, a, false, b2,
                                                     (short)0, acc2, false, false);
    }

    const float* nb = norms + (size_t)b * SDIM;
    float adj = 0.0f, dist = 0.0f;
    tcl_extract(acc0, c0, t0, half, nb, adj, dist);
    tcl_extract(acc1, c1, t0, half, nb, adj, dist);
    tcl_extract(acc2, c2, t0, half, nb, adj, dist);

    adj  = waveReduceSum(adj);
    dist = waveReduceSum(dist);
    if (lane == 0) {
        const float scaleAdj  = 1.0f / ((float)BDIM * (float)(SDIM - 1));
        const float scaleDist = 1.0f / ((float)BDIM * DIST_COUNT);
        atomicAdd(out, adj * scaleAdj + dist * scaleDist);
    }
}

extern "C" void kernel_launch(void* const* d_in, const int* in_sizes, int n_in,
                              void* d_out, int out_size, void* d_ws, size_t ws_size,
                              hipStream_t stream) {
    (void)in_sizes; (void)n_in; (void)out_size; (void)ws_size;
    const float* h   = (const float*)d_in[0];
    float* out       = (float*)d_out;
    float* norms     = (float*)d_ws;          // B*S floats = 256 KB scratch

    const int rows   = BDIM * SDIM;           // 65536
    const int wavesPerBlock = 8;              // 256 threads
    tcl_norms_kernel<<<rows / wavesPerBlock, 256, 0, stream>>>(h, norms, out);

    const int tiles = BDIM * (SDIM / 16);     // 4096 waves
    tcl_gram_kernel<<<tiles, 32, 0, stream>>>(h, norms, out);
}